// concat_40836549050799
// MI455X (gfx1250) — compile-verified
//
#include <hip/hip_runtime.h>
#include <hip/hip_bf16.h>

// Reference: x.reshape(B, 1, S*D) on a contiguous [B, S, D] fp32 tensor.
// Row-major layout is identical before and after the reshape, so the kernel
// is a pure device-to-device copy of 25,165,824 floats (~100.66 MB each way).
//
// MI455X roofline: 0 FLOPs, ~201 MB HBM traffic -> bandwidth-bound,
// floor ~8.6 us at 23.3 TB/s. Optimal path = coalesced 128-bit vector
// memory ops with non-temporal hints (dataset > 192 MB L2, zero reuse).

typedef float v4f __attribute__((ext_vector_type(4)));

__global__ void reshape_copy_f32x4(const v4f* __restrict__ in4,
                                   v4f* __restrict__ out4,
                                   long n4,
                                   const float* __restrict__ in_s,
                                   float* __restrict__ out_s,
                                   long n_total) {
  long i = (long)blockIdx.x * blockDim.x + threadIdx.x;
  long stride = (long)gridDim.x * blockDim.x;

  // Vectorized body: global_load_b128 / global_store_b128 with TH=NT.
  for (long v = i; v < n4; v += stride) {
    v4f val = __builtin_nontemporal_load(&in4[v]);
    __builtin_nontemporal_store(val, &out4[v]);
  }

  // Scalar tail (n_total not divisible by 4). For this problem the tail is
  // empty (25,165,824 % 4 == 0) so this loop body never executes, but it
  // keeps the kernel correct for arbitrary sizes.
  long tail_start = n4 * 4;
  for (long t = tail_start + i; t < n_total; t += stride) {
    float val = __builtin_nontemporal_load(&in_s[t]);
    __builtin_nontemporal_store(val, &out_s[t]);
  }
}

extern "C" void kernel_launch(void* const* d_in, const int* in_sizes, int n_in,
                              void* d_out, int out_size, void* d_ws, size_t ws_size,
                              hipStream_t stream) {
  (void)n_in; (void)d_ws; (void)ws_size; (void)out_size;

  const float* x = (const float*)d_in[0];
  float* out = (float*)d_out;

  long n_total = (long)in_sizes[0];   // 64 * 512 * 768 = 25,165,824
  long n4 = n_total / 4;              // 6,291,456 float4 transfers

  const int threads = 256;            // 8 wave32s per block
  // One float4 per thread in a single pass; grid-stride loop makes the
  // kernel robust if the grid is ever clamped.
  long blocks_l = (n4 + threads - 1) / threads;
  if (blocks_l < 1) blocks_l = 1;
  if (blocks_l > 1048576) blocks_l = 1048576;
  int blocks = (int)blocks_l;

  reshape_copy_f32x4<<<blocks, threads, 0, stream>>>(
      (const v4f*)x, (v4f*)out, n4, x, out, n_total);
}